// LandmarkMarkMatch_48344151884381
// MI455X (gfx1250) — compile-verified
//
#include <hip/hip_runtime.h>
#include <math.h>

// ---------------------------------------------------------------------------
// LandmarkMarkMatch for MI455X (gfx1250, wave32, WMMA).
// All f32 matmuls use V_WMMA_F32_16X16X4_F32 (exact fp32).
// Working set (~85MB) is L2-resident (192MB); compute-bound on f32 WMMA pipe.
// Top-k structural sparsity exploited exactly: kv/attention run on the 4096
// selected rows only; the 12288 zero rows contribute Z0*exp(-m) to the
// softmax denominator analytically.
// ---------------------------------------------------------------------------

typedef __attribute__((ext_vector_type(2))) float v2f;
typedef __attribute__((ext_vector_type(8))) float v8f;

#define N_TOT   16384
#define HEADS   8
#define DHEAD   64
#define KSEL    4096            // N_TOT * 0.25
#define ZMASS   12288.0f        // N_TOT - KSEL

#define ALDS_STRIDE 20          // 20 floats = 80B: 16B-aligned rows, 16 distinct banks

// ---------------------------------------------------------------------------
// f32 WMMA GEMM, compile-time specialized: C[M,N] = alpha * A' @ B' (+bias[n])
//   WG = 128 threads = 4 waves; WG tile 64(M) x 64(N); wave tile 16 x 64.
//   A panel [64 x 16k] staged in LDS via coalesced float4 loads; each A frag
//   feeds 4 WMMAs. B frags loaded direct (NT: contiguous b64 per lane;
//   NN: two 64B row segments per half). All dims divide exactly (no tails).
// ---------------------------------------------------------------------------
template <bool GATHER, bool TRANSB, bool HAS_BIAS>
__global__ void __launch_bounds__(128)
gemm_tile_kernel(const float* __restrict__ A, int lda,
                 const int* __restrict__ rowidx,
                 const float* __restrict__ B, int ldb,
                 const float* __restrict__ bias,
                 float alpha,
                 float* __restrict__ C, int ldc,
                 int K)
{
    __shared__ float Alds[64 * ALDS_STRIDE];
    __shared__ int   sidx[64];

    const int tid  = threadIdx.x;
    const int wave = tid >> 5;
    const int lane = tid & 31;
    const int half = lane >> 4;          // 0: K+0/1, 1: K+2/3 (frag halves)
    const int l16  = lane & 15;

    const int m0 = blockIdx.x * 64;
    const int n0 = blockIdx.y * 64;

    if (GATHER) {
        if (tid < 64) sidx[tid] = rowidx[m0 + tid];
        __syncthreads();
    }

    // staging assignment: thread -> rows (r0, r0+32), 4-float column chunk c0s
    const int r0  = tid >> 2;            // 0..31
    const int c0s = (tid & 3) * 4;       // 0,4,8,12
    const float* Ag0 = A + (size_t)(GATHER ? sidx[r0]      : (m0 + r0))      * lda + c0s;
    const float* Ag1 = A + (size_t)(GATHER ? sidx[r0 + 32] : (m0 + r0 + 32)) * lda + c0s;

    const float* arow_lds = &Alds[(wave * 16 + l16) * ALDS_STRIDE];

    v8f acc[4] = {};

    for (int kb = 0; kb < K; kb += 16) {
        __syncthreads();
        *(float4*)&Alds[ r0       * ALDS_STRIDE + c0s] = *(const float4*)(Ag0 + kb);
        *(float4*)&Alds[(r0 + 32) * ALDS_STRIDE + c0s] = *(const float4*)(Ag1 + kb);
        __syncthreads();

        #pragma unroll
        for (int kk = 0; kk < 16; kk += 4) {
            // A 16x4 fragment (ISA 7.12.2): lanes<16 -> K=kk,kk+1 ; lanes>=16 -> kk+2,kk+3
            v2f a;
            a.x = arow_lds[kk + 2 * half];
            a.y = arow_lds[kk + 2 * half + 1];
            #pragma unroll
            for (int ns = 0; ns < 4; ++ns) {
                v2f b;
                if (TRANSB) {            // B^T[k][n] = B[n][k] -> contiguous float2
                    const float* bp = B + (size_t)(n0 + ns * 16 + l16) * ldb
                                        + kb + kk + 2 * half;
                    b.x = bp[0];
                    b.y = bp[1];
                } else {                 // B[k][n] row-major
                    const float* bp = B + (size_t)(kb + kk + 2 * half) * ldb
                                        + n0 + ns * 16 + l16;
                    b.x = bp[0];
                    b.y = bp[ldb];
                }
                acc[ns] = __builtin_amdgcn_wmma_f32_16x16x4_f32(
                              false, a, false, b, (short)0, acc[ns], false, false);
            }
        }
    }

    #pragma unroll
    for (int ns = 0; ns < 4; ++ns) {
        const int  nc = n0 + ns * 16 + l16;
        const float bv = HAS_BIAS ? bias[nc] : 0.0f;
        #pragma unroll
        for (int r = 0; r < 8; ++r) {    // C/D layout: vgpr r -> M=r / r+8
            const int row = m0 + wave * 16 + r + 8 * half;
            C[(size_t)row * ldc + nc] = alpha * acc[ns][r] + bv;
        }
    }
}

// ---------------------------------------------------------------------------
// Row max over 512 columns, one wave per row (writes sim_M directly to d_out)
// ---------------------------------------------------------------------------
__global__ void __launch_bounds__(128)
rowmax_kernel(const float* __restrict__ L, float* __restrict__ out)
{
    const int wave = threadIdx.x >> 5, lane = threadIdx.x & 31;
    const int row  = blockIdx.x * 4 + wave;
    const float* Lr = L + (size_t)row * 512;
    float m = -INFINITY;
    #pragma unroll
    for (int j = 0; j < 16; ++j) m = fmaxf(m, Lr[lane + 32 * j]);
    #pragma unroll
    for (int mk = 16; mk; mk >>= 1) m = fmaxf(m, __shfl_xor(m, mk, 32));
    if (lane == 0) out[row] = m;
}

// ---------------------------------------------------------------------------
// Deterministic top-k (k=4096 of 16384): single-WG 4-bit radix select of the
// k-th largest key, then LDS-scan compaction. Ties broken by lowest index
// (matches jax.lax.top_k). Output: idx[k] (order arbitrary; feeds sums only).
// ---------------------------------------------------------------------------
__device__ __forceinline__ unsigned sim_key(float f) {
    unsigned u = __float_as_uint(f);
    return (u & 0x80000000u) ? ~u : (u | 0x80000000u);   // monotone in value
}

__global__ void __launch_bounds__(1024)
topk_kernel(const float* __restrict__ sim, int* __restrict__ idx)
{
    __shared__ unsigned hist[16];
    __shared__ unsigned sprefix;
    __shared__ int      sremaining;
    __shared__ unsigned scan[1024];
    __shared__ int      sGtot;

    const int tid = threadIdx.x;
    if (tid == 0) { sprefix = 0u; sremaining = KSEL; }
    __syncthreads();

    for (int d = 7; d >= 0; --d) {
        const int shift = d * 4;
        if (tid < 16) hist[tid] = 0u;
        __syncthreads();
        const unsigned pre = sprefix;
        for (int i = tid; i < N_TOT; i += 1024) {
            unsigned kk = sim_key(sim[i]);
            bool match = (d == 7) || ((kk >> (shift + 4)) == (pre >> (shift + 4)));
            if (match) atomicAdd(&hist[(kk >> shift) & 15u], 1u);
        }
        __syncthreads();
        if (tid == 0) {
            int rem = sremaining;
            unsigned p2 = sprefix;
            for (int v = 15; v >= 0; --v) {
                int c = (int)hist[v];
                if (rem > c) { rem -= c; }
                else { sprefix = p2 | ((unsigned)v << shift); sremaining = rem; break; }
            }
        }
        __syncthreads();
    }
    const unsigned T = sprefix;      // key of the k-th largest element
    const int rneed  = sremaining;   // how many ties (==T) to keep, lowest idx first

    // --- compaction: 16 consecutive elements per thread ---
    const int base = tid * 16;
    int cg = 0, ce = 0;
    #pragma unroll
    for (int j = 0; j < 16; ++j) {
        unsigned kk = sim_key(sim[base + j]);
        cg += (kk > T); ce += (kk == T);
    }

    // inclusive Hillis-Steele scan of cg
    scan[tid] = (unsigned)cg; __syncthreads();
    for (int off = 1; off < 1024; off <<= 1) {
        unsigned t = (tid >= off) ? scan[tid - off] : 0u;
        __syncthreads();
        scan[tid] += t;
        __syncthreads();
    }
    int off_g = (int)scan[tid] - cg;
    if (tid == 1023) sGtot = (int)scan[1023];
    __syncthreads();
    const int G = sGtot;             // #(key > T) == KSEL - rneed
    __syncthreads();

    // scan of ce (reuse buffer)
    scan[tid] = (unsigned)ce; __syncthreads();
    for (int off = 1; off < 1024; off <<= 1) {
        unsigned t = (tid >= off) ? scan[tid - off] : 0u;
        __syncthreads();
        scan[tid] += t;
        __syncthreads();
    }
    int off_e = (int)scan[tid] - ce;

    #pragma unroll
    for (int j = 0; j < 16; ++j) {
        unsigned kk = sim_key(sim[base + j]);
        if (kk > T) {
            idx[off_g++] = base + j;
        } else if (kk == T) {
            if (off_e < rneed) idx[G + off_e] = base + j;
            ++off_e;
        }
    }
}

// ---------------------------------------------------------------------------
// Flash attention over the 4096 compacted rows + analytic zero-mass term.
// One wave per (head, 16-row q tile). q tile lives in 16 A-frags (scaled by
// 1/8 once). S tile from 16 WMMAs; online softmax with shfl_xor row
// reductions; P re-laid out C->A via a per-wave LDS tile; P@V via WMMAs.
// ---------------------------------------------------------------------------
__global__ void __launch_bounds__(128)
flash_attn_kernel(const float* __restrict__ Q,    // [512, 512] (m, h*64+d)
                  const float* __restrict__ KV,   // [4096, 1024] k|v
                  float* __restrict__ O)          // [512, 512]
{
    __shared__ float pshare[4][16 * 17];
    const int wave = threadIdx.x >> 5, lane = threadIdx.x & 31;
    const int half = lane >> 4, l16 = lane & 15;
    const int h    = blockIdx.y;
    const int m0   = (blockIdx.x * 4 + wave) * 16;
    const int c0   = h * DHEAD;

    // resident q fragments (pre-scaled by dim_heads^-0.5 = 0.125)
    v2f qa[16];
    {
        const float* Qrow = Q + (size_t)(m0 + l16) * 512 + c0;
        #pragma unroll
        for (int c = 0; c < 16; ++c) {
            qa[c].x = Qrow[4 * c + 2 * half]     * 0.125f;
            qa[c].y = Qrow[4 * c + 2 * half + 1] * 0.125f;
        }
    }

    v8f o0 = {}, o1 = {}, o2 = {}, o3 = {};
    float m8[8], l8[8];
    #pragma unroll
    for (int r = 0; r < 8; ++r) { m8[r] = -INFINITY; l8[r] = 0.0f; }

    float* P = &pshare[wave][0];

    for (int s0 = 0; s0 < KSEL; s0 += 16) {
        // S = q @ k^T  (B^T[k=dh][n=s] = KV[s0+n][c0+dh])
        v8f acc = {};
        const float* Krow = KV + (size_t)(s0 + l16) * 1024 + c0;
        #pragma unroll
        for (int c = 0; c < 16; ++c) {
            v2f b;
            b.x = Krow[4 * c + 2 * half];
            b.y = Krow[4 * c + 2 * half + 1];
            acc = __builtin_amdgcn_wmma_f32_16x16x4_f32(false, qa[c], false, b,
                                                        (short)0, acc, false, false);
        }
        // online softmax per row (row = r + 8*half, cols striped over 16 lanes)
        #pragma unroll
        for (int r = 0; r < 8; ++r) {
            float sv = acc[r];
            float rm = sv;
            rm = fmaxf(rm, __shfl_xor(rm, 1, 32));
            rm = fmaxf(rm, __shfl_xor(rm, 2, 32));
            rm = fmaxf(rm, __shfl_xor(rm, 4, 32));
            rm = fmaxf(rm, __shfl_xor(rm, 8, 32));
            float mnew = fmaxf(m8[r], rm);
            float sc   = __expf(m8[r] - mnew);
            float p    = __expf(sv - mnew);
            float rs = p;
            rs += __shfl_xor(rs, 1, 32);
            rs += __shfl_xor(rs, 2, 32);
            rs += __shfl_xor(rs, 4, 32);
            rs += __shfl_xor(rs, 8, 32);
            l8[r] = l8[r] * sc + rs;
            m8[r] = mnew;
            o0[r] *= sc; o1[r] *= sc; o2[r] *= sc; o3[r] *= sc;
            P[(r + 8 * half) * 17 + l16] = p;     // row-major stash (pad 17)
        }
        // O += P @ V   (A-frags re-read from LDS; same-wave DS ops are in order)
        #pragma unroll
        for (int c = 0; c < 4; ++c) {
            v2f pa;
            pa.x = P[l16 * 17 + 4 * c + 2 * half];
            pa.y = P[l16 * 17 + 4 * c + 2 * half + 1];
            const float* V0 = KV + (size_t)(s0 + 4 * c + 2 * half) * 1024 + 512 + c0;
            const float* V1 = V0 + 1024;
            v2f b;
            b.x = V0[l16];      b.y = V1[l16];
            o0 = __builtin_amdgcn_wmma_f32_16x16x4_f32(false, pa, false, b, (short)0, o0, false, false);
            b.x = V0[16 + l16]; b.y = V1[16 + l16];
            o1 = __builtin_amdgcn_wmma_f32_16x16x4_f32(false, pa, false, b, (short)0, o1, false, false);
            b.x = V0[32 + l16]; b.y = V1[32 + l16];
            o2 = __builtin_amdgcn_wmma_f32_16x16x4_f32(false, pa, false, b, (short)0, o2, false, false);
            b.x = V0[48 + l16]; b.y = V1[48 + l16];
            o3 = __builtin_amdgcn_wmma_f32_16x16x4_f32(false, pa, false, b, (short)0, o3, false, false);
        }
    }

    // finalize: 12288 zero rows contribute ZMASS*exp(-m_f) to the denominator
    #pragma unroll
    for (int r = 0; r < 8; ++r) {
        float mf    = fmaxf(m8[r], 0.0f);
        float ex    = __expf(m8[r] - mf);
        float denom = l8[r] * ex + ZMASS * __expf(-mf);
        float inv   = ex / denom;
        float* Or = O + (size_t)(m0 + r + 8 * half) * 512 + c0;
        Or[l16]      = o0[r] * inv;
        Or[16 + l16] = o1[r] * inv;
        Or[32 + l16] = o2[r] * inv;
        Or[48 + l16] = o3[r] * inv;
    }
}

// ---------------------------------------------------------------------------
// M = LayerNorm(out + M), one 128-thread block per row of 512
// ---------------------------------------------------------------------------
__global__ void __launch_bounds__(128)
residual_ln_kernel(const float* __restrict__ X, float* __restrict__ Mb)
{
    __shared__ float red[128];
    const int row = blockIdx.x, tid = threadIdx.x;
    const float* Xr = X + (size_t)row * 512;
    float* Mr = Mb + (size_t)row * 512;

    float v[4], s = 0.0f;
    #pragma unroll
    for (int i = 0; i < 4; ++i) { v[i] = Xr[tid + 128 * i] + Mr[tid + 128 * i]; s += v[i]; }
    red[tid] = s; __syncthreads();
    for (int o = 64; o; o >>= 1) { if (tid < o) red[tid] += red[tid + o]; __syncthreads(); }
    const float mu = red[0] * (1.0f / 512.0f);
    __syncthreads();

    float s2 = 0.0f;
    #pragma unroll
    for (int i = 0; i < 4; ++i) { float d = v[i] - mu; s2 += d * d; }
    red[tid] = s2; __syncthreads();
    for (int o = 64; o; o >>= 1) { if (tid < o) red[tid] += red[tid + o]; __syncthreads(); }
    const float inv = rsqrtf(red[0] * (1.0f / 512.0f) + 1e-5f);
    #pragma unroll
    for (int i = 0; i < 4; ++i) Mr[tid + 128 * i] = (v[i] - mu) * inv;
}

// ---------------------------------------------------------------------------
// Softmax stats on cached final logits: overwrite L with p = exp(l - lse),
// emit per-block partials of (rowmax - lse) and lse^2. One wave per row.
// ---------------------------------------------------------------------------
__global__ void __launch_bounds__(128)
emb_stats_kernel(float* __restrict__ L, float* __restrict__ pl, float* __restrict__ pz)
{
    __shared__ float sl[4], sz[4];
    const int wave = threadIdx.x >> 5, lane = threadIdx.x & 31;
    const int row  = blockIdx.x * 4 + wave;
    float* Lr = L + (size_t)row * 512;

    float m = -INFINITY;
    #pragma unroll
    for (int j = 0; j < 16; ++j) m = fmaxf(m, Lr[lane + 32 * j]);
    #pragma unroll
    for (int mk = 16; mk; mk >>= 1) m = fmaxf(m, __shfl_xor(m, mk, 32));

    float s = 0.0f;
    #pragma unroll
    for (int j = 0; j < 16; ++j) s += __expf(Lr[lane + 32 * j] - m);
    #pragma unroll
    for (int mk = 16; mk; mk >>= 1) s += __shfl_xor(s, mk, 32);
    const float lse = m + __logf(s);

    #pragma unroll
    for (int j = 0; j < 16; ++j) Lr[lane + 32 * j] = __expf(Lr[lane + 32 * j] - lse);

    if (lane == 0) { sl[wave] = m - lse; sz[wave] = lse * lse; }
    __syncthreads();
    if (threadIdx.x == 0) {
        float a = 0.0f, b = 0.0f;
        #pragma unroll
        for (int w = 0; w < 4; ++w) { a += sl[w]; b += sz[w]; }
        pl[blockIdx.x] = a; pz[blockIdx.x] = b;
    }
}

__global__ void __launch_bounds__(256)
loss_reduce_kernel(const float* __restrict__ pl, const float* __restrict__ pz,
                   int n, float* __restrict__ out2)
{
    __shared__ float bl[256], bz[256];
    const int tid = threadIdx.x;
    float a = 0.0f, b = 0.0f;
    for (int i = tid; i < n; i += 256) { a += pl[i]; b += pz[i]; }
    bl[tid] = a; bz[tid] = b; __syncthreads();
    for (int s = 128; s; s >>= 1) {
        if (tid < s) { bl[tid] += bl[tid + s]; bz[tid] += bz[tid + s]; }
        __syncthreads();
    }
    if (tid == 0) {
        out2[0] = -(bl[0] * (1.0f / (float)N_TOT));   // result_loss
        out2[1] =   bz[0] * (1.0f / (float)N_TOT);    // zloss
    }
}

// ---------------------------------------------------------------------------
// Orchestration
// ---------------------------------------------------------------------------
extern "C" void kernel_launch(void* const* d_in, const int* in_sizes, int n_in,
                              void* d_out, int out_size, void* d_ws, size_t ws_size,
                              hipStream_t stream)
{
    (void)in_sizes; (void)n_in; (void)out_size; (void)ws_size;

    const float* data   = (const float*)d_in[0];   // [16384, 512]
    const float* Memory = (const float*)d_in[1];   // [512, 512]
    const float* Wkv    = (const float*)d_in[2];   // [512, 1024]
    const float* Wq     = (const float*)d_in[3];   // [2, 512, 512]
    const float* Wm     = (const float*)d_in[4];   // [2, 512, 512]
    const float* bm     = (const float*)d_in[5];   // [2, 512]
    // d_in[6] = istest (traced as 0 -> training path)

    float* out     = (float*)d_out;
    float* sim     = out;                                    // [16384]
    float* out_ebs = out + N_TOT;                            // [16384*512]
    float* losses  = out + N_TOT + (size_t)N_TOT * 512;      // [2]

    float* ws     = (float*)d_ws;                  // ~54.6 MB total
    float* logits = ws;                            // 16384*512
    float* kv     = logits + (size_t)N_TOT * 512;  // 4096*1024
    float* qbuf   = kv + (size_t)KSEL * 1024;      // 512*512
    float* obuf   = qbuf + 512 * 512;              // 512*512
    float* outbuf = obuf + 512 * 512;              // 512*512
    float* Mbuf   = outbuf + 512 * 512;            // 512*512
    float* pl     = Mbuf + 512 * 512;              // 4096
    float* pz     = pl + 4096;                     // 4096
    int*   idx    = (int*)(pz + 4096);             // 4096

    hipMemcpyAsync(Mbuf, Memory, 512 * 512 * sizeof(float),
                   hipMemcpyDeviceToDevice, stream);

    const dim3 blk(128);
    const dim3 gSim(N_TOT / 64, 512 / 64);   // [16384,512] tiles
    const dim3 gSq(512 / 64, 512 / 64);      // 512x512x512
    const dim3 gKv(KSEL / 64, 1024 / 64);    // 4096x1024x512
    const float scale = 0.125f;              // dim_heads^-0.5

    // initial sim + selection
    gemm_tile_kernel<false, true, false><<<gSim, blk, 0, stream>>>(
        data, 512, nullptr, Mbuf, 512, nullptr, scale, logits, 512, 512);
    rowmax_kernel<<<N_TOT / 4, 128, 0, stream>>>(logits, sim);
    topk_kernel<<<1, 1024, 0, stream>>>(sim, idx);

    for (int outer = 0; outer < 2; ++outer) {
        for (int l = 0; l < 2; ++l) {
            const float* Wql = Wq + (size_t)l * 512 * 512;
            const float* Wml = Wm + (size_t)l * 512 * 512;
            const float* bml = bm + (size_t)l * 512;
            // q = M @ Wq_l
            gemm_tile_kernel<false, false, false><<<gSq, blk, 0, stream>>>(
                Mbuf, 512, nullptr, Wql, 512, nullptr, 1.0f, qbuf, 512, 512);
            // kv = gather(data, idx) @ Wkv   (only the 4096 nonzero rows)
            gemm_tile_kernel<true, false, false><<<gKv, blk, 0, stream>>>(
                data, 512, idx, Wkv, 1024, nullptr, 1.0f, kv, 1024, 512);
            // o = softmax(q k^T / 8) v  with analytic zero-row mass
            flash_attn_kernel<<<dim3(8, HEADS), blk, 0, stream>>>(qbuf, kv, obuf);
            // out = o @ Wm_l + bm_l
            gemm_tile_kernel<false, false, true><<<gSq, blk, 0, stream>>>(
                obuf, 512, nullptr, Wml, 512, bml, 1.0f, outbuf, 512, 512);
            // M = LN(out + M)
            residual_ln_kernel<<<512, 128, 0, stream>>>(outbuf, Mbuf);
        }
        // sim refresh (logits cached for calucate_emb after final iteration)
        gemm_tile_kernel<false, true, false><<<gSim, blk, 0, stream>>>(
            data, 512, nullptr, Mbuf, 512, nullptr, scale, logits, 512, 512);
        rowmax_kernel<<<N_TOT / 4, 128, 0, stream>>>(logits, sim);
        if (outer == 0) topk_kernel<<<1, 1024, 0, stream>>>(sim, idx);
        // (selection after the last iteration is dead in the reference)
    }

    // calucate_emb on cached logits: stats + in-place p, then out_ebs = p @ M
    emb_stats_kernel<<<N_TOT / 4, 128, 0, stream>>>(logits, pl, pz);
    loss_reduce_kernel<<<1, 256, 0, stream>>>(pl, pz, N_TOT / 4, losses);
    gemm_tile_kernel<false, false, false><<<gSim, blk, 0, stream>>>(
        logits, 512, nullptr, Mbuf, 512, nullptr, 1.0f, out_ebs, 512, 512);
}